// RecurrentLinearSelfAttention_5420248728194
// MI455X (gfx1250) — compile-verified
//
#include <hip/hip_runtime.h>

typedef __attribute__((ext_vector_type(2))) float v2f;
typedef __attribute__((ext_vector_type(4))) float v4f;
typedef __attribute__((ext_vector_type(8))) float v8f;

#define CDIM 128
#define NTOKB 1024          // tokens per batch (H*W)
#define BATCH 4
#define NTOK  (BATCH * NTOKB)   // 4096

__device__ __forceinline__ float fmap(float x) {
    // elu(x) + 1 : x>0 ? x+1 : exp(x)
    return x > 0.0f ? x + 1.0f : __expf(x);
}

// ---------------------------------------------------------------------------
// Kernel 1: Y = W @ F projections via V_WMMA_F32_16X16X4_F32 (fp32 WMMA).
// One wave per 16x16 tile of (Cout x N). B-fragment (feature) reused across
// wq/wk/wv. Epilogue: +bias, feature-map for q/k, transpose-store to (B,N,C).
// ---------------------------------------------------------------------------
__global__ __launch_bounds__(256) void proj_gemm_wmma(
    const float* __restrict__ feat,
    const float* __restrict__ wq, const float* __restrict__ bq,
    const float* __restrict__ wk, const float* __restrict__ bk,
    const float* __restrict__ wv, const float* __restrict__ bv,
    float* __restrict__ Qw, float* __restrict__ Kw, float* __restrict__ Vw)
{
    const int lane = threadIdx.x & 31;
    const int wid  = threadIdx.x >> 5;
    const int tile = blockIdx.x * 8 + wid;      // 0..2047
    const int b    = tile >> 9;                 // / (64 n-tiles * 8 c-tiles)
    const int rem  = tile & 511;
    const int n0   = (rem >> 3) << 4;           // token-tile base
    const int c0   = (rem & 7)  << 4;           // cout-tile base
    const int hi   = lane >> 4, lo = lane & 15;

    const float* Fb = feat + (size_t)b * CDIM * NTOKB;   // F[ci][n] row-major

    v8f aq = {}; v8f ak = {}; v8f av = {};

    // A-matrix 16x4 f32 layout: lane row M=lo; VGPR0/1 hold K = 2*hi, 2*hi+1
    const int arow = (c0 + lo) * CDIM;

    for (int k0 = 0; k0 < CDIM; k0 += 4) {
        const int kb = k0 + 2 * hi;
        v2f a_q = *(const v2f*)(wq + arow + kb);
        v2f a_k = *(const v2f*)(wk + arow + kb);
        v2f a_v = *(const v2f*)(wv + arow + kb);
        // B-matrix 4x16 f32: row K striped across lanes within a VGPR
        v2f bm;
        bm.x = Fb[(size_t)kb * NTOKB + n0 + lo];
        bm.y = Fb[(size_t)(kb + 1) * NTOKB + n0 + lo];
        if (k0 + 4 < CDIM)
            __builtin_prefetch(Fb + (size_t)(kb + 4) * NTOKB + n0 + lo, 0, 1);

        aq = __builtin_amdgcn_wmma_f32_16x16x4_f32(false, a_q, false, bm,
                                                   (short)0, aq, false, false);
        ak = __builtin_amdgcn_wmma_f32_16x16x4_f32(false, a_k, false, bm,
                                                   (short)0, ak, false, false);
        av = __builtin_amdgcn_wmma_f32_16x16x4_f32(false, a_v, false, bm,
                                                   (short)0, av, false, false);
    }

    // D 16x16 f32: VGPR r -> (M = r + 8*hi, N = lo)
    const int    n    = n0 + lo;
    const size_t base = ((size_t)b * NTOKB + n) * CDIM;
#pragma unroll
    for (int r = 0; r < 8; ++r) {
        const int    co  = c0 + r + 8 * hi;
        const size_t idx = base + co;
        Qw[idx] = fmap(aq[r] + bq[co]);
        Kw[idx] = fmap(ak[r] + bk[co]);
        Vw[idx] = av[r] + bv[co];
    }
}

// ---------------------------------------------------------------------------
// Kernel 2: per-token epilogue. One wave per token; lane t owns m = 4t..4t+3.
//   s = Q.K (wave32 shuffle reduction)
//   RLSA = s/(s+eps) * V ; Si = K ; Ri[c][m] = K[c]*V[m] streamed NT.
// Ri is 256 MB -> the kernel is pure store bandwidth; every Ri store is a
// coalesced 512B/wave non-temporal b128.
// ---------------------------------------------------------------------------
__global__ __launch_bounds__(256) void lsa_outer(
    const float* __restrict__ Qw, const float* __restrict__ Kw,
    const float* __restrict__ Vw, float* __restrict__ out)
{
    const int lane  = threadIdx.x & 31;
    const int wid   = threadIdx.x >> 5;
    const int token = blockIdx.x * 8 + wid;     // 0..4095

    __shared__ float kbuf[8][CDIM];

    const size_t tb = (size_t)token * CDIM;
    v4f q4 = *(const v4f*)(Qw + tb + 4 * lane);
    v4f k4 = *(const v4f*)(Kw + tb + 4 * lane);
    v4f v4 = *(const v4f*)(Vw + tb + 4 * lane);

    *(v4f*)(&kbuf[wid][4 * lane]) = k4;
    __syncthreads();

    float p = q4.x * k4.x + q4.y * k4.y + q4.z * k4.z + q4.w * k4.w;
#pragma unroll
    for (int off = 16; off > 0; off >>= 1)
        p += __shfl_xor(p, off, 32);
    const float zs = p / (p + 1e-6f);           // (Q.K)/(Q.K+eps)

    float* rlsa = out;                                            // NTOK*C
    float* ri   = out + (size_t)NTOK * CDIM;                      // NTOK*C*C
    float* si   = out + (size_t)NTOK * CDIM * (CDIM + 1);         // NTOK*C

    v4f o4;
    o4.x = zs * v4.x; o4.y = zs * v4.y; o4.z = zs * v4.z; o4.w = zs * v4.w;
    *(v4f*)(rlsa + tb + 4 * lane) = o4;
    *(v4f*)(si   + tb + 4 * lane) = k4;

    float* rbase = ri + (size_t)token * CDIM * CDIM + 4 * lane;
#pragma unroll 4
    for (int c = 0; c < CDIM; ++c) {
        const float kc = kbuf[wid][c];
        v4f r4;
        r4.x = kc * v4.x; r4.y = kc * v4.y;
        r4.z = kc * v4.z; r4.w = kc * v4.w;
        __builtin_nontemporal_store(r4, (v4f*)(rbase + (size_t)c * CDIM));
    }
}

extern "C" void kernel_launch(void* const* d_in, const int* in_sizes, int n_in,
                              void* d_out, int out_size, void* d_ws, size_t ws_size,
                              hipStream_t stream)
{
    const float* feat = (const float*)d_in[0];
    const float* wq   = (const float*)d_in[1];
    const float* bq   = (const float*)d_in[2];
    const float* wk   = (const float*)d_in[3];
    const float* bk   = (const float*)d_in[4];
    const float* wv   = (const float*)d_in[5];
    const float* bv   = (const float*)d_in[6];

    float* ws = (float*)d_ws;                   // 3 * NTOK * C floats = 6 MB
    float* Qw = ws;
    float* Kw = ws + (size_t)NTOK * CDIM;
    float* Vw = ws + 2 * (size_t)NTOK * CDIM;

    // 2048 wave-tiles / 8 waves per block
    proj_gemm_wmma<<<256, 256, 0, stream>>>(feat, wq, bq, wk, bk, wv, bv,
                                            Qw, Kw, Vw);
    // 4096 tokens / 8 waves per block
    lsa_outer<<<512, 256, 0, stream>>>(Qw, Kw, Vw, (float*)d_out);
}